// SS2Dv0_63531156242553
// MI455X (gfx1250) — compile-verified
//
#include <hip/hip_runtime.h>
#include <hip/hip_bf16.h>

typedef float v2f __attribute__((ext_vector_type(2)));
typedef float v8f __attribute__((ext_vector_type(8)));

#define BB   8
#define HH   64
#define WW   64
#define LL   4096
#define DM   96
#define DI   192
#define NST  16
#define RNK  6
#define KGRP 4
#define CPROJ 38           // RNK + 2*NST
#define ML   (BB * LL)     // 32768 token rows

#define WMMA_F32X4(A, Bf, C) \
  __builtin_amdgcn_wmma_f32_16x16x4_f32(false, (A), false, (Bf), (short)0, (C), false, false)

// CDNA5 async global->LDS path (ASYNCcnt). Guarded so either toolchain compiles.
// Builtin signature (from hipcc diagnostic): param0 = global int* (AS1), so we
// cast through non-const int pointers in the right address spaces.
#if defined(__AMDGCN__) && __has_builtin(__builtin_amdgcn_global_load_async_to_lds_b32) && \
    __has_builtin(__builtin_amdgcn_s_wait_asynccnt)
#define HAVE_ASYNC_LDS 1
typedef __attribute__((address_space(1))) int async_gint_t;
typedef __attribute__((address_space(3))) int async_lint_t;
#endif

__device__ __forceinline__ float silu(float v) {
  return v * (1.0f / (1.0f + __expf(-v)));
}

// Map scan position t of direction k to HW-order linear index l.
// k bit1 = reversed, k bit0 = WH (transposed) order.
__device__ __forceinline__ int scan_to_l(int k, int t) {
  int tt = (k & 2) ? (LL - 1 - t) : t;
  if (k & 1) return ((tt & (HH - 1)) << 6) | (tt >> 6);  // t = w*H + h -> l = h*W + w
  return tt;
}

// ---------------------------------------------------------------------------
// K0: zero the y accumulator
// ---------------------------------------------------------------------------
__global__ void k_zero(float* __restrict__ p, int n) {
  int i = blockIdx.x * 256 + threadIdx.x;
  if (i < n) p[i] = 0.0f;
}

// ---------------------------------------------------------------------------
// K1: xz = x @ Win^T ; z = silu(z) stored [B,L,Di]; xc = silu(xh) stored [B,Di,L]
// WMMA fp32 GEMM: M = 32768, N = 384, K = 96
// ---------------------------------------------------------------------------
__global__ __launch_bounds__(128) void k_inproj(const float* __restrict__ x,
                                                const float* __restrict__ w,
                                                float* __restrict__ xc,
                                                float* __restrict__ z) {
  const int lane = threadIdx.x & 31;
  const int wave = threadIdx.x >> 5;
  const int tile = blockIdx.x * 4 + wave;
  const int NT = (2 * DI) / 16;            // 24 col tiles
  const int ntile = tile % NT;
  const int mtile = tile / NT;
  const int half = lane >> 4;
  const int l15 = lane & 15;
  const int m = mtile * 16 + l15;          // A row for this lane
  const int cb = ntile * 16 + l15;         // B col for this lane

  const float* __restrict__ arow = x + (size_t)m * DM;
  const float* __restrict__ brow = w + (size_t)cb * DM;  // W[c, j]
  v8f acc = {};
  for (int kk = 0; kk < DM; kk += 4) {
    const int k0 = kk + half * 2;
    v2f a, bf;
    a[0] = arow[k0];     a[1] = arow[k0 + 1];
    bf[0] = brow[k0];    bf[1] = brow[k0 + 1];
    acc = WMMA_F32X4(a, bf, acc);
  }
#pragma unroll
  for (int r = 0; r < 8; ++r) {
    const int mr = mtile * 16 + r + 8 * half;
    const int b = mr >> 12;
    const int l = mr & (LL - 1);
    const float s = silu(acc[r]);
    if (cb < DI) xc[((size_t)b * DI + cb) * LL + l] = s;
    else         z[((size_t)b * LL + l) * DI + (cb - DI)] = s;
  }
}

// ---------------------------------------------------------------------------
// K2: x_dbl = Wx[k] (38x192, padded to 48) @ xs[b,k] (192 x L, gathered from xc)
// Splits rows into dt_raw [B,K,6,L], Bs [B,K,16,L], Cs [B,K,16,L]
// ---------------------------------------------------------------------------
__global__ __launch_bounds__(128) void k_xdbl(const float* __restrict__ xc,
                                              const float* __restrict__ wx,
                                              float* __restrict__ dtr,
                                              float* __restrict__ Bsd,
                                              float* __restrict__ Csd) {
  const int lane = threadIdx.x & 31;
  const int wave = threadIdx.x >> 5;
  const int tile = blockIdx.x * 4 + wave;
  const int TPK = 3 * (LL / 16);           // 768 tiles per (b,k)
  const int bk = tile / TPK;
  const int rem = tile % TPK;
  const int ct = rem / (LL / 16);          // 0..2 (c row-tile, padded to 48)
  const int tt = rem % (LL / 16);          // 0..255 (t col-tile)
  const int b = bk >> 2, k = bk & 3;
  const int half = lane >> 4;
  const int l15 = lane & 15;
  const int c = ct * 16 + l15;
  const int t = tt * 16 + l15;
  const int l = scan_to_l(k, t);

  const float* __restrict__ wxk = wx + (size_t)k * CPROJ * DI;
  const float* __restrict__ xcb = xc + (size_t)b * DI * LL + l;
  v8f acc = {};
  for (int dd = 0; dd < DI; dd += 4) {
    const int d0 = dd + half * 2;
    v2f a, bf;
    const bool cv = (c < CPROJ);
    a[0] = cv ? wxk[(size_t)c * DI + d0]     : 0.0f;
    a[1] = cv ? wxk[(size_t)c * DI + d0 + 1] : 0.0f;
    bf[0] = xcb[(size_t)d0 * LL];
    bf[1] = xcb[(size_t)(d0 + 1) * LL];
    acc = WMMA_F32X4(a, bf, acc);
  }
#pragma unroll
  for (int r = 0; r < 8; ++r) {
    const int cr = ct * 16 + r + 8 * half;
    const int tc = tt * 16 + l15;
    const float v = acc[r];
    if (cr < RNK)
      dtr[((size_t)bk * RNK + cr) * LL + tc] = v;
    else if (cr < RNK + NST)
      Bsd[((size_t)bk * NST + (cr - RNK)) * LL + tc] = v;
    else if (cr < CPROJ)
      Csd[((size_t)bk * NST + (cr - RNK - NST)) * LL + tc] = v;
  }
}

// ---------------------------------------------------------------------------
// K3: selective scan. One block per (b,k); 192 threads = 1 channel each.
// dt_proj + softplus fused (W_dt row in registers). Shared (dtr,B,C) streams
// double-buffered through LDS with CDNA5 async global->LDS loads (ASYNCcnt):
// chunk i+1 is in flight while chunk i computes. Width padded to 39 so every
// thread issues exactly 13 async loads/chunk -> s_wait_asynccnt(13) retires
// precisely the current chunk (per-wave in-order completion).
// y merged across directions with float atomics into HW-order buffer.
// ---------------------------------------------------------------------------
#define CHUNK 64
#define SW 39                        // 6 dtr + 16 B + 16 C + 1 pad
#define LPT 13                       // CHUNK*SW / 192 == 13 exactly
__global__ __launch_bounds__(192) void k_scan(const float* __restrict__ xc,
                                              const float* __restrict__ dtr,
                                              const float* __restrict__ Bsd,
                                              const float* __restrict__ Csd,
                                              const float* __restrict__ wdt,
                                              const float* __restrict__ dtb,
                                              const float* __restrict__ Alog,
                                              const float* __restrict__ Ds,
                                              float* __restrict__ yhw) {
  __shared__ float stage[2][CHUNK][SW];

  const int bk = blockIdx.x;
  const int b = bk >> 2, k = bk & 3;
  const int d = threadIdx.x;
  const int kd = k * DI + d;

  float wrow[RNK];
#pragma unroll
  for (int r = 0; r < RNK; ++r) wrow[r] = wdt[(size_t)kd * RNK + r];
  const float bias = dtb[kd];
  float Arow[NST];
#pragma unroll
  for (int n = 0; n < NST; ++n) Arow[n] = -__expf(Alog[(size_t)kd * NST + n]);
  const float Dd = Ds[kd];

  float h[NST];
#pragma unroll
  for (int n = 0; n < NST; ++n) h[n] = 0.0f;

  const float* __restrict__ xcbd   = xc  + ((size_t)b * DI + d) * LL;
  float* __restrict__ ybd          = yhw + ((size_t)b * DI + d) * LL;
  const float* __restrict__ dtr_bk = dtr + (size_t)bk * RNK * LL;
  const float* __restrict__ B_bk   = Bsd + (size_t)bk * NST * LL;
  const float* __restrict__ C_bk   = Csd + (size_t)bk * NST * LL;

  // Issue one chunk's worth of stream data (uniform LPT loads per thread).
  auto issue_chunk = [&](int buf, int t0) {
#pragma unroll 1
    for (int j = 0; j < LPT; ++j) {
      const int idx = threadIdx.x + j * 192;
      const int tt = idx / SW, cc = idx % SW;
      const int t = t0 + tt;
      const float* src;
      if (cc < RNK)                 src = &dtr_bk[(size_t)cc * LL + t];
      else if (cc < RNK + NST)      src = &B_bk[(size_t)(cc - RNK) * LL + t];
      else if (cc < RNK + 2 * NST)  src = &C_bk[(size_t)(cc - RNK - NST) * LL + t];
      else                          src = &dtr_bk[t];   // pad: keep count uniform
      float* dst = &stage[buf][tt][cc];
#ifdef HAVE_ASYNC_LDS
      __builtin_amdgcn_global_load_async_to_lds_b32((async_gint_t*)(float*)src,
                                                    (async_lint_t*)dst, 0, 0);
#else
      *dst = *src;
#endif
    }
  };

  issue_chunk(0, 0);
  int buf = 0;
  for (int t0 = 0; t0 < LL; t0 += CHUNK) {
    if (t0 + CHUNK < LL) {
      issue_chunk(buf ^ 1, t0 + CHUNK);      // overlap next chunk with compute
#ifdef HAVE_ASYNC_LDS
      __builtin_amdgcn_s_wait_asynccnt(LPT); // retire current chunk only
#endif
      __builtin_prefetch(&xcbd[scan_to_l(k, t0 + CHUNK)], 0, 1);
    } else {
#ifdef HAVE_ASYNC_LDS
      __builtin_amdgcn_s_wait_asynccnt(0);
#endif
    }
    __syncthreads();                         // all waves' LDS writes visible

    for (int tt = 0; tt < CHUNK; ++tt) {
      const int t = t0 + tt;
      const int l = scan_to_l(k, t);
      const float xv = xcbd[l];
      float acc = bias;
#pragma unroll
      for (int r = 0; r < RNK; ++r) acc = fmaf(wrow[r], stage[buf][tt][r], acc);
      const float delta = (acc > 20.0f) ? acc : __logf(1.0f + __expf(acc));
      const float dx = delta * xv;
      float y = Dd * xv;                     // D-skip term (Ds input)
#pragma unroll
      for (int n = 0; n < NST; ++n) {
        h[n] = h[n] * __expf(delta * Arow[n]) + dx * stage[buf][tt][RNK + n];
        y = fmaf(h[n], stage[buf][tt][RNK + NST + n], y);
      }
      atomicAdd(&ybd[l], y);                 // merge 4 directions in HW order
    }
    __syncthreads();                         // readers done before buffer reuse
    buf ^= 1;
  }
}

// ---------------------------------------------------------------------------
// K4: out = silu( (y * silu(z)) @ Wout^T )   M=32768, N=96, K=192 (fp32 WMMA)
// ---------------------------------------------------------------------------
__global__ __launch_bounds__(128) void k_out(const float* __restrict__ yhw,
                                             const float* __restrict__ z,
                                             const float* __restrict__ wo,
                                             float* __restrict__ out) {
  const int lane = threadIdx.x & 31;
  const int wave = threadIdx.x >> 5;
  const int tile = blockIdx.x * 4 + wave;
  const int NT = DM / 16;                  // 6 col tiles
  const int ntile = tile % NT;
  const int mtile = tile / NT;
  const int half = lane >> 4;
  const int l15 = lane & 15;
  const int m = mtile * 16 + l15;
  const int b = m >> 12;
  const int l = m & (LL - 1);
  const int cb = ntile * 16 + l15;

  const float* __restrict__ yb   = yhw + (size_t)b * DI * LL + l;   // [d] stride LL
  const float* __restrict__ zrow = z + (size_t)m * DI;
  const float* __restrict__ wrow = wo + (size_t)cb * DI;            // Wout[c, d]
  v8f acc = {};
  for (int dd = 0; dd < DI; dd += 4) {
    const int d0 = dd + half * 2;
    v2f a, bf;
    a[0] = yb[(size_t)d0 * LL] * zrow[d0];
    a[1] = yb[(size_t)(d0 + 1) * LL] * zrow[d0 + 1];
    bf[0] = wrow[d0];
    bf[1] = wrow[d0 + 1];
    acc = WMMA_F32X4(a, bf, acc);
  }
#pragma unroll
  for (int r = 0; r < 8; ++r) {
    const int mr = mtile * 16 + r + 8 * half;
    out[(size_t)mr * DM + cb] = silu(acc[r]);
  }
}

// ---------------------------------------------------------------------------
extern "C" void kernel_launch(void* const* d_in, const int* in_sizes, int n_in,
                              void* d_out, int out_size, void* d_ws, size_t ws_size,
                              hipStream_t stream) {
  const float* x    = (const float*)d_in[0];  // [8,64,64,96]
  const float* wip  = (const float*)d_in[1];  // [384,96]
  const float* wx   = (const float*)d_in[2];  // [4,38,192]
  const float* wdt  = (const float*)d_in[3];  // [4,192,6]
  const float* dtb  = (const float*)d_in[4];  // [4,192]
  const float* alog = (const float*)d_in[5];  // [768,16]
  const float* ds   = (const float*)d_in[6];  // [768]
  const float* wo   = (const float*)d_in[7];  // [96,192]
  float* out = (float*)d_out;

  // Workspace layout (floats)
  float* ws  = (float*)d_ws;
  const size_t SZ_XC = (size_t)BB * DI * LL;          // 6,291,456
  const size_t SZ_Z  = SZ_XC;
  const size_t SZ_DT = (size_t)BB * KGRP * RNK * LL;  //   786,432
  const size_t SZ_BS = (size_t)BB * KGRP * NST * LL;  // 2,097,152
  float* xc  = ws;
  float* z   = xc + SZ_XC;
  float* dtr = z + SZ_Z;
  float* Bsd = dtr + SZ_DT;
  float* Csd = Bsd + SZ_BS;
  float* yhw = Csd + SZ_BS;

  // K0: zero y accumulator
  {
    const int n = (int)SZ_XC;
    k_zero<<<(n + 255) / 256, 256, 0, stream>>>(yhw, n);
  }
  // K1: in_proj (2048 m-tiles * 24 n-tiles, 4 waves/block)
  k_inproj<<<(ML / 16) * (2 * DI / 16) / 4, 128, 0, stream>>>(x, wip, xc, z);
  // K2: x_dbl projections (32 bk * 3 c-tiles * 256 t-tiles, 4 waves/block)
  k_xdbl<<<(BB * KGRP) * 3 * (LL / 16) / 4, 128, 0, stream>>>(xc, wx, dtr, Bsd, Csd);
  // K3: selective scan, one block per (b,k)
  k_scan<<<BB * KGRP, DI, 0, stream>>>(xc, dtr, Bsd, Csd, wdt, dtb, alog, ds, yhw);
  // K4: gate + out_proj (2048 m-tiles * 6 n-tiles, 4 waves/block)
  k_out<<<(ML / 16) * (DM / 16) / 4, 128, 0, stream>>>(yhw, z, wo, out);
}